// QueryEncoder_1185410973872
// MI455X (gfx1250) — compile-verified
//
#include <hip/hip_runtime.h>
#include <stdint.h>

namespace {
constexpr int kB       = 512;
constexpr int kS       = 256;
constexpr int kV       = 50257;
constexpr int kPad     = 1;
constexpr int kWords   = (kV + 31) / 32;            // 1571 dwords of presence bits
constexpr int kThreads = 512;                        // 16 wave32s per block
constexpr int kSplit   = 4;                          // V-chunks per row
constexpr int kChunk   = (kV + kSplit - 1) / kSplit; // 12565
}

typedef __attribute__((ext_vector_type(4))) float f32x4;
typedef __attribute__((address_space(1))) int gint_t;   // global int
typedef __attribute__((address_space(3))) int lint_t;   // LDS int

#if defined(__gfx1250__) &&                                         \
    __has_builtin(__builtin_amdgcn_global_load_async_to_lds_b32) && \
    __has_builtin(__builtin_amdgcn_s_wait_asynccnt)
#define MH_ASYNC_LDS 1
#else
#define MH_ASYNC_LDS 0
#endif

__global__ __launch_bounds__(kThreads)
void multihot_rowmask(const int* __restrict__ ids,   // [B,S] token ids (i32)
                      const float* __restrict__ w,   // [V]
                      float* __restrict__ out)       // [B,V]
{
  __shared__ unsigned int bitmap[kWords + 1];        // +1 guard word for 64-bit window reads
  __shared__ int sids[kS];

  const int b   = blockIdx.x;
  const int tid = threadIdx.x;

  // Zero the presence bitmap (ds stores).
  for (int i = tid; i < kWords + 1; i += kThreads) bitmap[i] = 0u;

  // Stage this row's 256 ids into LDS. On gfx1250, use the async
  // global->LDS copy tracked by ASYNCcnt; otherwise plain load+ds_store.
#if MH_ASYNC_LDS
  if (tid < kS) {
    __builtin_amdgcn_global_load_async_to_lds_b32(
        (gint_t*)(ids + (size_t)b * kS + tid),
        (lint_t*)(sids + tid),
        0, 0);
  }
  __builtin_amdgcn_s_wait_asynccnt(0);
#else
  if (tid < kS) sids[tid] = ids[(size_t)b * kS + tid];
#endif
  __syncthreads();

  // Set presence bits with LDS atomics (ds_or_b32). Skipping PAD here is
  // equivalent to the reference's zeroing of the PAD column afterwards.
  if (tid < kS) {
    const unsigned int v = (unsigned int)sids[tid];
    if (v != (unsigned int)kPad)
      atomicOr(&bitmap[v >> 5], 1u << (v & 31u));
  }
  __syncthreads();

  // ---- Stream this block's V-chunk of the output row, one write pass ----
  const int vlo = blockIdx.y * kChunk;
  const int vhi = (vlo + kChunk < kV) ? (vlo + kChunk) : kV;

  float* __restrict__ orow = out + (size_t)b * kV;

  // V is odd, so row starts drift mod 16B: peel to the first v >= vlo where
  // the global float index (b*kV + v) is 16B aligned, then do b128 NT stores.
  const int mis = (int)(((size_t)b * (size_t)kV + (size_t)vlo) & 3);
  int valign    = vlo + ((4 - mis) & 3);
  if (valign > vhi) valign = vhi;
  const int nvec  = (vhi - valign) >> 2;
  const int vtail = valign + (nvec << 2);

  // Scalar head (<= 3 elements).
  for (int v = vlo + tid; v < valign; v += kThreads) {
    const float val = ((bitmap[v >> 5] >> (v & 31)) & 1u) ? w[v] : 0.0f;
    __builtin_nontemporal_store(val, orow + v);
  }
  // Scalar tail (<= 3 elements).
  for (int v = vtail + tid; v < vhi; v += kThreads) {
    const float val = ((bitmap[v >> 5] >> (v & 31)) & 1u) ? w[v] : 0.0f;
    __builtin_nontemporal_store(val, orow + v);
  }
  // Vector body: aligned 16B non-temporal stores; presence bits read through
  // a 64-bit window since v0 is not necessarily a multiple of 4 mod 32.
  for (int g = tid; g < nvec; g += kThreads) {
    const int v0 = valign + (g << 2);
    const int wi = v0 >> 5;
    const unsigned long long win =
        (unsigned long long)bitmap[wi] |
        ((unsigned long long)bitmap[wi + 1] << 32);
    const unsigned int bits = (unsigned int)(win >> (v0 & 31)) & 0xFu;
    f32x4 o;
    o.x = (bits & 1u) ? w[v0 + 0] : 0.0f;
    o.y = (bits & 2u) ? w[v0 + 1] : 0.0f;
    o.z = (bits & 4u) ? w[v0 + 2] : 0.0f;
    o.w = (bits & 8u) ? w[v0 + 3] : 0.0f;
    __builtin_nontemporal_store(o, (f32x4*)(orow + v0));
  }
}

extern "C" void kernel_launch(void* const* d_in, const int* in_sizes, int n_in,
                              void* d_out, int out_size, void* d_ws, size_t ws_size,
                              hipStream_t stream) {
  (void)in_sizes; (void)n_in; (void)out_size; (void)d_ws; (void)ws_size;
  const int*   ids = (const int*)d_in[0];   // input_ids [B,S]
  const float* w   = (const float*)d_in[1]; // weights  [V]
  float*       out = (float*)d_out;         // [B,V] f32

  dim3 grid(kB, kSplit, 1);
  multihot_rowmask<<<grid, dim3(kThreads, 1, 1), 0, stream>>>(ids, w, out);
}